// GraphSAGEConv_26087631356317
// MI455X (gfx1250) — compile-verified
//
#include <hip/hip_runtime.h>

typedef __attribute__((ext_vector_type(2))) float v2f;
typedef __attribute__((ext_vector_type(8))) float v8f;

#define F_IN 128
#define F2   256
#define LDSS 260     // padded LDS row stride (floats): conflict-free, 16B-aligned
#define RPB  32      // rows per block in GEMM

// ---------------- kernel 1: zero edge-count histogram ----------------
__global__ void sage_zero_i(int* __restrict__ p, int n) {
    int i = blockIdx.x * blockDim.x + threadIdx.x;
    if (i < n) p[i] = 0;
}

// ---------------- kernel 2: degree histogram (int atomics) ----------------
__global__ void sage_hist(const int* __restrict__ dst, int* __restrict__ counts, int E) {
    int e = blockIdx.x * blockDim.x + threadIdx.x;
    if (e < E) atomicAdd(&counts[dst[e]], 1);
}

// ---------------- kernel 3: exclusive prefix scan (single block) ----------
__global__ void __launch_bounds__(256)
sage_scan(const int* __restrict__ counts, int* __restrict__ offsets,
          int* __restrict__ cursor, int N) {
    __shared__ int psum[257];
    int tid = threadIdx.x;
    int per = (N + 255) / 256;
    int s = tid * per;
    int e = s + per; if (e > N) e = N;
    int sum = 0;
    for (int i = s; i < e; ++i) sum += counts[i];
    psum[tid] = sum;
    __syncthreads();
    if (tid == 0) {
        int run = 0;
        for (int i = 0; i < 256; ++i) { int t = psum[i]; psum[i] = run; run += t; }
        psum[256] = run;
    }
    __syncthreads();
    int run = psum[tid];
    for (int i = s; i < e; ++i) {
        offsets[i] = run;
        cursor[i]  = run;
        run += counts[i];
    }
    if (e == N) offsets[N] = psum[256];
}

// ---------------- kernel 4: bucket-scatter edge sources by dst ------------
__global__ void sage_sort(const int* __restrict__ src, const int* __restrict__ dst,
                          int* __restrict__ cursor, int* __restrict__ sortedSrc, int E) {
    int e = blockIdx.x * blockDim.x + threadIdx.x;
    if (e < E) {
        int pos = atomicAdd(&cursor[dst[e]], 1);
        sortedSrc[pos] = src[e];
    }
}

// ---------------- kernel 5: per-node mean aggregation (no f32 atomics) ----
// one wave per node; lane handles 4 contiguous features (float4 gather)
__global__ void __launch_bounds__(256)
sage_aggregate(const float* __restrict__ x, const int* __restrict__ sortedSrc,
               const int* __restrict__ offsets, float* __restrict__ agg, int N) {
    int v    = (blockIdx.x * 256 + threadIdx.x) >> 5;
    int lane = threadIdx.x & 31;
    if (v >= N) return;
    int s = offsets[v], e = offsets[v + 1];
    float4 acc = {0.f, 0.f, 0.f, 0.f};
    for (int i = s; i < e; ++i) {
        int u = sortedSrc[i];
        float4 t = ((const float4*)(x + (size_t)u * F_IN))[lane];
        acc.x += t.x; acc.y += t.y; acc.z += t.z; acc.w += t.w;
    }
    float dg = (float)(e - s);
    if (dg < 1.0f) dg = 1.0f;
    float4 r = {acc.x / dg, acc.y / dg, acc.z / dg, acc.w / dg};
    ((float4*)(agg + (size_t)v * F_IN))[lane] = r;
}

// ---------------- kernel 6: WMMA GEMM + ReLU ------------------------------
// block = 256 threads = 8 waves = 2 row-groups x 4 col-groups.
// each wave: 16 rows x 32 cols (2 accumulators share one A fetch).
__global__ void __launch_bounds__(256)
sage_gemm(const float* __restrict__ x,
          const float* __restrict__ agg,
          const float* __restrict__ W,     // [128, 256] row-major
          float* __restrict__ out,         // [N, 128]
          int N) {
    __shared__ float ah[RPB * LDSS];

    const int tid     = threadIdx.x;
    const int wave    = tid >> 5;
    const int lane    = tid & 31;
    const int rowBase = blockIdx.x * RPB;

    // stage 32x256 A-tile (x | agg) as pure float4 copies
    for (int idx = tid; idx < RPB * (F2 / 4); idx += 256) {
        int r  = idx >> 6;       // 64 float4 per row
        int c4 = idx & 63;
        int g = rowBase + r;
        if (g >= N) g = N - 1;   // clamp (rows unused if OOB)
        float4 v;
        if (c4 < 32) v = ((const float4*)(x   + (size_t)g * F_IN))[c4];
        else         v = ((const float4*)(agg + (size_t)g * F_IN))[c4 - 32];
        *(float4*)(&ah[r * LDSS + c4 * 4]) = v;
    }
    __syncthreads();

    const int rg   = wave >> 2;          // row group 0..1
    const int cg   = wave & 3;           // col group 0..3
    const int colBase = cg * 32;
    const int mr   = lane & 15;          // A row (M) / B col (N)
    const int koff = (lane >> 4) * 2;    // lanes 16-31 hold K+2,K+3

    const float* w0   = W + (size_t)(colBase + mr) * F2 + koff;
    const float* w1   = w0 + (size_t)16 * F2;
    const float* arow = &ah[(rg * 16 + mr) * LDSS + koff];

    v8f c0 = {0.f, 0.f, 0.f, 0.f, 0.f, 0.f, 0.f, 0.f};
    v8f c1 = {0.f, 0.f, 0.f, 0.f, 0.f, 0.f, 0.f, 0.f};

#pragma unroll 4
    for (int k = 0; k < F2; k += 4) {
        v2f a  = *(const v2f*)(arow + k);   // ds_load_b64 (shared by both tiles)
        v2f b0 = *(const v2f*)(w0 + k);     // global_load_b64 (L2-resident W)
        v2f b1 = *(const v2f*)(w1 + k);
        c0 = __builtin_amdgcn_wmma_f32_16x16x4_f32(false, a, false, b0,
                                                   (short)0, c0, false, false);
        c1 = __builtin_amdgcn_wmma_f32_16x16x4_f32(false, a, false, b1,
                                                   (short)0, c1, false, false);
    }

    // D layout: VGPR r -> row (r + 8*(lane>=16)), col = lane&15 ; ReLU
    const int orow = rowBase + rg * 16 + (lane >> 4) * 8;
    float* ob = out + (size_t)orow * F_IN + colBase + mr;

    if (rowBase + RPB <= N) {
#pragma unroll
        for (int r = 0; r < 8; ++r) {
            float v0 = c0[r], v1 = c1[r];
            ob[(size_t)r * F_IN]      = v0 > 0.f ? v0 : 0.f;
            ob[(size_t)r * F_IN + 16] = v1 > 0.f ? v1 : 0.f;
        }
    } else {
#pragma unroll
        for (int r = 0; r < 8; ++r) {
            if (orow + r < N) {
                float v0 = c0[r], v1 = c1[r];
                ob[(size_t)r * F_IN]      = v0 > 0.f ? v0 : 0.f;
                ob[(size_t)r * F_IN + 16] = v1 > 0.f ? v1 : 0.f;
            }
        }
    }
}

// ---------------- launcher ----------------
extern "C" void kernel_launch(void* const* d_in, const int* in_sizes, int n_in,
                              void* d_out, int out_size, void* d_ws, size_t ws_size,
                              hipStream_t stream) {
    const float* x   = (const float*)d_in[0];
    const int*   src = (const int*)  d_in[1];
    const int*   dst = (const int*)  d_in[2];
    // d_in[3] = num_nodes scalar (device); use in_sizes instead
    const float* W   = (const float*)d_in[4];
    float* out = (float*)d_out;

    const int N = in_sizes[0] / F_IN;   // 50000
    const int E = in_sizes[1];          // 800000

    // workspace layout
    float* agg       = (float*)d_ws;                       // N*128 f32
    int*   counts    = (int*)(agg + (size_t)N * F_IN);     // N
    int*   offsets   = counts + N;                         // N+1
    int*   cursor    = offsets + (N + 1);                  // N
    int*   sortedSrc = cursor + N;                         // E

    sage_zero_i   <<<(N + 255) / 256, 256, 0, stream>>>(counts, N);
    sage_hist     <<<(E + 255) / 256, 256, 0, stream>>>(dst, counts, E);
    sage_scan     <<<1, 256, 0, stream>>>(counts, offsets, cursor, N);
    sage_sort     <<<(E + 255) / 256, 256, 0, stream>>>(src, dst, cursor, sortedSrc, E);

    int aggBlocks = (N * 32 + 255) / 256;     // one wave per node
    sage_aggregate<<<aggBlocks, 256, 0, stream>>>(x, sortedSrc, offsets, agg, N);

    sage_gemm     <<<(N + RPB - 1) / RPB, 256, 0, stream>>>(x, agg, W, out, N);
}